// RPNLoss_88716844466682
// MI455X (gfx1250) — compile-verified
//
#include <hip/hip_runtime.h>

// RPN loss (classification CE + attenuated regression) for MI455X / gfx1250.
// Stage 1: gather + per-lane accumulate + wave32 butterfly + per-block partials.
// Stage 2: single-block WMMA (v_wmma_f32_16x16x4_f32, B = ones) reduction of the
//          2048 block partials per quantity, then the scalar epilogue.

typedef float v2f __attribute__((ext_vector_type(2)));
typedef float v8f __attribute__((ext_vector_type(8)));

#define EPS_VAR 1e-3f
#define NB  2048   // blocks in stage 1 == partials per quantity (multiple of 64)
#define TPB 256

__global__ __launch_bounds__(TPB) void rpn_partial_kernel(
    const float* __restrict__ pred, const float* __restrict__ unc,
    const float* __restrict__ cls,  const float* __restrict__ gt,
    const int* __restrict__ label,  const int* __restrict__ vi,
    float* __restrict__ partials, int V)
{
    float cls_s = 0.f, pos_s = 0.f, neg_s = 0.f;
    int   pos_c = 0;
    const int stride = gridDim.x * blockDim.x;

    for (int i = blockIdx.x * blockDim.x + threadIdx.x; i < V; i += stride) {
        if (i + stride < V) __builtin_prefetch(&vi[i + stride], 0, 3);

        const int v   = vi[i];
        const int lab = label[v];

        // 2-class log-softmax CE: -(x_lab - lse) = (max - x_lab) + log1p(exp(-|d|))
        const float2 lg = ((const float2*)cls)[v];
        const float  xl = (lab == 1) ? lg.y : lg.x;
        cls_s += (fmaxf(lg.x, lg.y) - xl) + log1pf(expf(-fabsf(lg.x - lg.y)));

        const float4 p = ((const float4*)pred)[v];
        const float4 u = ((const float4*)unc)[v];
        const float4 g = ((const float4*)gt)[v];

        const float va = EPS_VAR + u.x * u.x, vb = EPS_VAR + u.y * u.y;
        const float vc = EPS_VAR + u.z * u.z, vd = EPS_VAR + u.w * u.w;
        const float ra = 1.f / va, rb = 1.f / vb, rc = 1.f / vc, rd = 1.f / vd;
        const float da = p.x - g.x, db = p.y - g.y, dc = p.z - g.z, dd = p.w - g.w;

        // sum(0.5*log(v_i)) == 0.5*log(prod v_i): 1 transcendental instead of 4
        const float pterm = 0.5f * (da*da*ra + db*db*rb + dc*dc*rc + dd*dd*rd)
                          + 0.5f * logf(va * vb * vc * vd);
        const float nterm = ra + rb + rc + rd;

        const bool ispos = (lab == 1);
        pos_s += ispos ? pterm : 0.f;
        neg_s += ispos ? 0.f   : nterm;
        pos_c += ispos ? 1     : 0;
    }

    // wave32 butterfly reduction
    for (int off = 16; off > 0; off >>= 1) {
        cls_s += __shfl_xor(cls_s, off);
        pos_s += __shfl_xor(pos_s, off);
        neg_s += __shfl_xor(neg_s, off);
        pos_c += __shfl_xor(pos_c, off);
    }

    __shared__ float sh[4][TPB / 32];
    const int lane = threadIdx.x & 31, wav = threadIdx.x >> 5;
    if (lane == 0) {
        sh[0][wav] = cls_s; sh[1][wav] = pos_s;
        sh[2][wav] = neg_s; sh[3][wav] = (float)pos_c;
    }
    __syncthreads();
    if (threadIdx.x < 4) {
        float t = 0.f;
        for (int w = 0; w < TPB / 32; ++w) t += sh[threadIdx.x][w];
        // quantity-major layout so stage 2 streams 8B-aligned float2 chunks
        partials[threadIdx.x * NB + blockIdx.x] = t;
    }
}

// Stage 2: 4 waves, one per quantity {class_sum, pos_sum, neg_sum, pos_cnt}.
// D += A_chunk x B_ones : D[m][n] = rowsum_m, so sum(all 256 D elems) = 16*total.
// Layout-agnostic: any bijection of 64 chunk values into A is valid with B==1.
__global__ __launch_bounds__(128) void rpn_final_kernel(
    const float* __restrict__ partials, float* __restrict__ out, int V)
{
    __shared__ float tot[4];
    const int wav  = threadIdx.x >> 5;   // 0..3, wave-uniform
    const int lane = threadIdx.x & 31;

    const float* P = partials + wav * NB;
    v8f c = {0.f, 0.f, 0.f, 0.f, 0.f, 0.f, 0.f, 0.f};
    const v2f ones = {1.0f, 1.0f};

    for (int ch = 0; ch < NB; ch += 64) {
        v2f a = *(const v2f*)(P + ch + 2 * lane);   // 64 consecutive partials/wave
        c = __builtin_amdgcn_wmma_f32_16x16x4_f32(
                /*neg_a=*/false, a, /*neg_b=*/false, ones,
                /*c_mod=*/(short)0, c, /*reuse_a=*/false, /*reuse_b=*/false);
    }

    float s = c[0] + c[1] + c[2] + c[3] + c[4] + c[5] + c[6] + c[7];
    for (int off = 1; off < 32; off <<= 1) s += __shfl_xor(s, off);
    if (lane == 0) tot[wav] = s * 0.0625f;          // /16: exact (pow2)
    __syncthreads();

    if (threadIdx.x == 0) {
        const float class_loss = tot[0] / (float)V;
        const float pos_cnt = tot[3];
        const float neg_cnt = (float)V - pos_cnt;   // counts partition V exactly
        float reg = 0.f;
        if (pos_cnt != 0.f && neg_cnt != 0.f)
            reg = tot[1] / fmaxf(pos_cnt, 1.f) + tot[2] / fmaxf(neg_cnt, 1.f);
        out[0] = class_loss + reg;
    }
}

extern "C" void kernel_launch(void* const* d_in, const int* in_sizes, int n_in,
                              void* d_out, int out_size, void* d_ws, size_t ws_size,
                              hipStream_t stream) {
    (void)n_in; (void)out_size; (void)ws_size;
    const float* pred  = (const float*)d_in[0];
    const float* unc   = (const float*)d_in[1];
    const float* cls   = (const float*)d_in[2];
    const float* gt    = (const float*)d_in[3];
    const int*   label = (const int*)d_in[4];
    const int*   vi    = (const int*)d_in[5];
    const int V = in_sizes[5];

    float* partials = (float*)d_ws;   // 4 * NB floats = 32 KB, rewritten every call

    rpn_partial_kernel<<<NB, TPB, 0, stream>>>(pred, unc, cls, gt, label, vi,
                                               partials, V);
    rpn_final_kernel<<<1, 128, 0, stream>>>(partials, (float*)d_out, V);
}